// ShipPhysics_1529008358106
// MI455X (gfx1250) — compile-verified
//
#include <hip/hip_runtime.h>
#include <math.h>

// ---------------------------------------------------------------------------
// ShipPhysics: 1M independent ships, RK45 (Dormand-Prince) x 10 substeps.
// Element-wise, VALU/transcendental bound (AI ~32 FLOP/B). No matmul structure
// -> WMMA not applicable. CDNA5 paths used: async global->LDS staging of the
// 22 table entries (ASYNCcnt + s_wait_asynccnt), wave32 sizing, native
// v_sqrt/v_rsq/v_sin/v_cos.
// ---------------------------------------------------------------------------

#define GLOBAL_AS __attribute__((address_space(1)))
#define LDS_AS    __attribute__((address_space(3)))

#if __has_builtin(__builtin_amdgcn_global_load_async_to_lds_b32)
#define HAVE_ASYNC_LDS 1
#else
#define HAVE_ASYNC_LDS 0
#endif

namespace shipk {

constexpr int   kNSUB = 10;
constexpr float kDT   = 0.02f;

// Dormand-Prince coefficients pre-multiplied by h = DT/NSUB (double-precision
// products rounded once to fp32).
constexpr double Hd = 0.02 / 10.0;
constexpr float c21 = (float)(Hd * (1.0 / 5.0));
constexpr float c31 = (float)(Hd * (3.0 / 40.0));
constexpr float c32 = (float)(Hd * (9.0 / 40.0));
constexpr float c41 = (float)(Hd * (44.0 / 45.0));
constexpr float c42 = (float)(Hd * (-56.0 / 15.0));
constexpr float c43 = (float)(Hd * (32.0 / 9.0));
constexpr float c51 = (float)(Hd * (19372.0 / 6561.0));
constexpr float c52 = (float)(Hd * (-25360.0 / 2187.0));
constexpr float c53 = (float)(Hd * (64448.0 / 6561.0));
constexpr float c54 = (float)(Hd * (-212.0 / 729.0));
constexpr float c61 = (float)(Hd * (9017.0 / 3168.0));
constexpr float c62 = (float)(Hd * (-355.0 / 33.0));
constexpr float c63 = (float)(Hd * (46732.0 / 5247.0));
constexpr float c64 = (float)(Hd * (49.0 / 176.0));
constexpr float c65 = (float)(Hd * (-5103.0 / 18656.0));
constexpr float b1  = (float)(Hd * (35.0 / 384.0));
constexpr float b3  = (float)(Hd * (500.0 / 1113.0));
constexpr float b4  = (float)(Hd * (125.0 / 192.0));
constexpr float b5  = (float)(Hd * (-2187.0 / 6784.0));
constexpr float b6  = (float)(Hd * (11.0 / 84.0));

// LDS table layout: turn[0..7] energy[8..11] drag[12..15] lift[16..17] thrust[18..21]
constexpr int kTabN = 22;

__device__ __forceinline__ void f_eval(float vr, float vi,
                                       float tfr, float tfi,
                                       float drag, float lift,
                                       float& ar, float& ai) {
    // s = |v|; a = thrust - drag*v*s -/+ lift*perp(v)*s
    float s  = __builtin_amdgcn_sqrtf(fmaf(vr, vr, vi * vi));  // v_sqrt_f32
    float ds = drag * s;
    float ls = lift * s;
    ar = tfr - ds * vr - ls * vi;
    ai = tfi - ds * vi + ls * vr;
}

}  // namespace shipk

__global__ __launch_bounds__(256) void ship_physics_kernel(
    const int*   __restrict__ actions,      // (N,5) int32
    const float* __restrict__ pos_r,
    const float* __restrict__ pos_i,
    const float* __restrict__ vel_r,
    const float* __restrict__ vel_i,
    const float* __restrict__ turn_off_in,
    const float* __restrict__ boost_in,
    const float* __restrict__ max_boost,
    const float* __restrict__ thrust_in,
    const float* __restrict__ turn_table,   // 8
    const float* __restrict__ energy_table, // 4
    const float* __restrict__ drag_table,   // 4
    const float* __restrict__ lift_table,   // 2
    const float* __restrict__ thrust_table, // 4
    float*       __restrict__ out,          // (8,N)
    int n) {
    using namespace shipk;

    __shared__ float tabs[kTabN];

    // --- Stage the 22 table floats into LDS (gfx1250 async global->LDS) -----
    const int t = (int)threadIdx.x;
    if (t < kTabN) {
        const float* src;
        int off;
        if (t < 8)        { src = turn_table;   off = t;      }
        else if (t < 12)  { src = energy_table; off = t - 8;  }
        else if (t < 16)  { src = drag_table;   off = t - 12; }
        else if (t < 18)  { src = lift_table;   off = t - 16; }
        else              { src = thrust_table; off = t - 18; }
#if HAVE_ASYNC_LDS
        __builtin_amdgcn_global_load_async_to_lds_b32(
            (GLOBAL_AS int*)(src + off), (LDS_AS int*)&tabs[t], 0, 0);
#else
        tabs[t] = src[off];
#endif
    }
#if HAVE_ASYNC_LDS
#if __has_builtin(__builtin_amdgcn_s_wait_asynccnt)
    __builtin_amdgcn_s_wait_asynccnt(0);
#else
    asm volatile("s_wait_asynccnt 0" ::: "memory");
#endif
#endif
    __syncthreads();

    const int i = (int)(blockIdx.x * blockDim.x + threadIdx.x);
    if (i >= n) return;

    // --- Coalesced input loads ---------------------------------------------
    const size_t abase = (size_t)i * 5;
    const int a0 = actions[abase + 0];
    const int a1 = actions[abase + 1];
    const int a2 = actions[abase + 2];
    const int a3 = actions[abase + 3];
    const int a4 = actions[abase + 4];

    const bool left  = a0 > 0;
    const bool right = a1 > 0;
    const int  sharp = a4;
    const int  idx_turn = a0 + 2 * a1 + 4 * sharp;
    const int  idx_fb   = a2 + 2 * a3;

    float toff = turn_off_in[i];
    const float turn_ang = tabs[idx_turn];            // ds_load_b32
    toff = (left && right) ? toff : turn_ang;

    const float co = __cosf(toff);                    // v_cos_f32
    const float si = __sinf(toff);                    // v_sin_f32

    float vr = vel_r[i];
    float vi = vel_i[i];
    float pr = pos_r[i];
    float pi = pos_i[i];

    // attitude at t0: normalize(v) rotated by toff
    const float inv0 = __builtin_amdgcn_rsqf(fmaf(vr, vr, vi * vi));  // v_rsq_f32
    const float dr0 = vr * inv0, di0 = vi * inv0;
    const float att_r = dr0 * co - di0 * si;
    const float att_i = dr0 * si + di0 * co;

    // boost / thrust
    const float cost = tabs[8 + idx_fb];
    float b = boost_in[i] - cost * kDT;
    b = fminf(fmaxf(b, 0.0f), max_boost[i]);
    float tm = tabs[18 + idx_fb];
    tm = (b > 0.0f) ? tm : 1.0f;
    const float th  = thrust_in[i];
    const float tfr = th * tm * att_r;
    const float tfi = th * tm * att_i;

    // drag / lift
    const int   turning = (left || right) ? 1 : 0;
    const float drag    = tabs[12 + turning + 2 * sharp];
    const float blift   = tabs[16 + sharp];
    const float tdir    = (right ? 1.0f : 0.0f) - (left ? 1.0f : 0.0f);
    const float lift    = (left != right) ? tdir * blift : 0.0f;

    // --- RK45 (Dormand-Prince), NSUB substeps, all in registers -------------
    for (int s = 0; s < kNSUB; ++s) {
        float k1ar, k1ai;
        f_eval(vr, vi, tfr, tfi, drag, lift, k1ar, k1ai);

        const float v2r = fmaf(c21, k1ar, vr);
        const float v2i = fmaf(c21, k1ai, vi);
        float k2ar, k2ai;
        f_eval(v2r, v2i, tfr, tfi, drag, lift, k2ar, k2ai);

        const float v3r = fmaf(c31, k1ar, fmaf(c32, k2ar, vr));
        const float v3i = fmaf(c31, k1ai, fmaf(c32, k2ai, vi));
        float k3ar, k3ai;
        f_eval(v3r, v3i, tfr, tfi, drag, lift, k3ar, k3ai);

        const float v4r = fmaf(c41, k1ar, fmaf(c42, k2ar, fmaf(c43, k3ar, vr)));
        const float v4i = fmaf(c41, k1ai, fmaf(c42, k2ai, fmaf(c43, k3ai, vi)));
        float k4ar, k4ai;
        f_eval(v4r, v4i, tfr, tfi, drag, lift, k4ar, k4ai);

        const float v5r = fmaf(c51, k1ar, fmaf(c52, k2ar, fmaf(c53, k3ar, fmaf(c54, k4ar, vr))));
        const float v5i = fmaf(c51, k1ai, fmaf(c52, k2ai, fmaf(c53, k3ai, fmaf(c54, k4ai, vi))));
        float k5ar, k5ai;
        f_eval(v5r, v5i, tfr, tfi, drag, lift, k5ar, k5ai);

        const float v6r = fmaf(c61, k1ar, fmaf(c62, k2ar, fmaf(c63, k3ar, fmaf(c64, k4ar, fmaf(c65, k5ar, vr)))));
        const float v6i = fmaf(c61, k1ai, fmaf(c62, k2ai, fmaf(c63, k3ai, fmaf(c64, k4ai, fmaf(c65, k5ai, vi)))));
        float k6ar, k6ai;
        f_eval(v6r, v6i, tfr, tfi, drag, lift, k6ar, k6ai);

        // dp/dt = v : B-weighted stage velocities; dv/dt = a : B-weighted accels
        const float npr = fmaf(b1, vr,   fmaf(b3, v3r,  fmaf(b4, v4r,  fmaf(b5, v5r,  fmaf(b6, v6r,  pr)))));
        const float npi = fmaf(b1, vi,   fmaf(b3, v3i,  fmaf(b4, v4i,  fmaf(b5, v5i,  fmaf(b6, v6i,  pi)))));
        const float nvr = fmaf(b1, k1ar, fmaf(b3, k3ar, fmaf(b4, k4ar, fmaf(b5, k5ar, fmaf(b6, k6ar, vr)))));
        const float nvi = fmaf(b1, k1ai, fmaf(b3, k3ai, fmaf(b4, k4ai, fmaf(b5, k5ai, fmaf(b6, k6ai, vi)))));
        pr = npr; pi = npi; vr = nvr; vi = nvi;
    }

    // --- Final attitude ------------------------------------------------------
    const float inv1 = __builtin_amdgcn_rsqf(fmaf(vr, vr, vi * vi));
    const float dr1 = vr * inv1, di1 = vi * inv1;
    const float ar1 = dr1 * co - di1 * si;
    const float ai1 = dr1 * si + di1 * co;

    // --- Field-major (8, N) coalesced stores --------------------------------
    const size_t N = (size_t)n;
    const size_t ii = (size_t)i;
    out[0 * N + ii] = pr;
    out[1 * N + ii] = pi;
    out[2 * N + ii] = vr;
    out[3 * N + ii] = vi;
    out[4 * N + ii] = ar1;
    out[5 * N + ii] = ai1;
    out[6 * N + ii] = b;
    out[7 * N + ii] = toff;
}

extern "C" void kernel_launch(void* const* d_in, const int* in_sizes, int n_in,
                              void* d_out, int out_size, void* d_ws, size_t ws_size,
                              hipStream_t stream) {
    (void)n_in; (void)d_ws; (void)ws_size; (void)out_size;
    const int*   actions      = (const int*)  d_in[0];
    const float* pos_r        = (const float*)d_in[1];
    const float* pos_i        = (const float*)d_in[2];
    const float* vel_r        = (const float*)d_in[3];
    const float* vel_i        = (const float*)d_in[4];
    const float* turn_offset  = (const float*)d_in[5];
    const float* boost        = (const float*)d_in[6];
    const float* max_boost    = (const float*)d_in[7];
    const float* thrust       = (const float*)d_in[8];
    const float* turn_table   = (const float*)d_in[9];
    const float* energy_table = (const float*)d_in[10];
    const float* drag_table   = (const float*)d_in[11];
    const float* lift_table   = (const float*)d_in[12];
    const float* thrust_table = (const float*)d_in[13];

    const int n = in_sizes[1];  // pos_r element count == N
    const int block = 256;      // 8 wave32 per block
    const int grid  = (n + block - 1) / block;

    ship_physics_kernel<<<grid, block, 0, stream>>>(
        actions, pos_r, pos_i, vel_r, vel_i, turn_offset, boost, max_boost,
        thrust, turn_table, energy_table, drag_table, lift_table, thrust_table,
        (float*)d_out, n);
}